// WholeNeuralEncoderDecoder_24885040513650
// MI455X (gfx1250) — compile-verified
//
#include <hip/hip_runtime.h>
#include <hip/hip_bf16.h>
#include <math.h>

// MI455X / gfx1250, wave32. FP32 WMMA path: V_WMMA_F32_16X16X4_F32.
typedef __attribute__((ext_vector_type(2))) float v2f;
typedef __attribute__((ext_vector_type(8))) float v8f;

#define NN  1024   // N nodes
#define DIN 512    // input feature dim
#define DD  128    // hidden dim

// ---------------------------------------------------------------------------
// v[m] = sum_l W2[m,l] * w3[l],  m in [0,256), l in [0,128)
// Tiny: one block of 256 threads.
// ---------------------------------------------------------------------------
__global__ void compute_v_kernel(const float* __restrict__ W2,
                                 const float* __restrict__ w3,
                                 float* __restrict__ v) {
    int m = threadIdx.x;                 // 0..255
    const float* row = W2 + (size_t)m * DD;
    float s = 0.f;
    #pragma unroll 8
    for (int l = 0; l < DD; ++l) s += row[l] * w3[l];
    v[m] = s;
}

// ---------------------------------------------------------------------------
// Generic row-major FP32 WMMA GEMM: C(MxN) = A(MxK) @ B(KxN).
// One wave per 16x16 output tile; K consumed 4 at a time by
// v_wmma_f32_16x16x4_f32. A-frag: lane holds A[M=lane&15, K=kb+2*hi .. +1]
// (contiguous -> float2 load). B-frag: lane holds B[kb+2*hi][N=lane&15] and
// B[kb+2*hi+1][...] (two coalesced strided loads).
// ---------------------------------------------------------------------------
__global__ void gemm_f32_wmma(const float* __restrict__ A,
                              const float* __restrict__ B,
                              float* __restrict__ C,
                              int M, int N, int K) {
    const int gwave  = (blockIdx.x * blockDim.x + threadIdx.x) >> 5;
    const int lane   = threadIdx.x & 31;
    const int tilesN = N >> 4;
    const int tm = gwave / tilesN;
    const int tn = gwave - tm * tilesN;
    const int lo = lane & 15;
    const int hi = lane >> 4;

    const float* arow = A + (size_t)(tm * 16 + lo) * K + 2 * hi;
    const float* bcol = B + (size_t)(2 * hi) * N + tn * 16 + lo;

    v8f acc = {};
    #pragma unroll 4
    for (int kb = 0; kb < K; kb += 4) {
        v2f af = *(const v2f*)(arow + kb);
        v2f bf;
        const float* bp = bcol + (size_t)kb * N;
        bf.x = bp[0];
        bf.y = bp[N];
        acc = __builtin_amdgcn_wmma_f32_16x16x4_f32(
            /*neg_a=*/false, af, /*neg_b=*/false, bf,
            /*c_mod=*/(short)0, acc, /*reuse_a=*/false, /*reuse_b=*/false);
    }

    // C/D layout: VGPR g -> row (g + 8*hi), col lo within the tile.
    float* crow = C + (size_t)(tm * 16 + hi * 8) * N + tn * 16 + lo;
    #pragma unroll
    for (int g = 0; g < 8; ++g)
        crow[(size_t)g * N] = acc[g];
}

// ---------------------------------------------------------------------------
// Per-row prep over z (1024 x 128): one wave per row.
//   zw[i,k] = z[i,k] * w3[128+k]
//   a[i]    = sum_k relu(z[i,k]) * v[k]
//   b[i]    = sum_k relu(z[i,k]) * v[128+k]
// Lane handles 4 consecutive k (float4), then wave-wide xor-shuffle reduce.
// ---------------------------------------------------------------------------
__global__ void row_stats_kernel(const float* __restrict__ z,
                                 const float* __restrict__ v,
                                 const float* __restrict__ w3,
                                 float* __restrict__ zw,
                                 float* __restrict__ av,
                                 float* __restrict__ bv) {
    const int row  = (blockIdx.x * blockDim.x + threadIdx.x) >> 5;
    const int lane = threadIdx.x & 31;
    const int k0   = lane * 4;

    const float4 zr  = *(const float4*)(z  + (size_t)row * DD + k0);
    const float4 w3e = *(const float4*)(w3 + DD + k0);
    const float4 va  = *(const float4*)(v  + k0);
    const float4 vb  = *(const float4*)(v  + DD + k0);

    float4 zwv;
    zwv.x = zr.x * w3e.x; zwv.y = zr.y * w3e.y;
    zwv.z = zr.z * w3e.z; zwv.w = zr.w * w3e.w;
    *(float4*)(zw + (size_t)row * DD + k0) = zwv;

    const float rx = fmaxf(zr.x, 0.f), ry = fmaxf(zr.y, 0.f);
    const float rz = fmaxf(zr.z, 0.f), rw = fmaxf(zr.w, 0.f);
    float ra = rx * va.x + ry * va.y + rz * va.z + rw * va.w;
    float rb = rx * vb.x + ry * vb.y + rz * vb.z + rw * vb.w;

    #pragma unroll
    for (int off = 16; off > 0; off >>= 1) {
        ra += __shfl_xor(ra, off, 32);
        rb += __shfl_xor(rb, off, 32);
    }
    if (lane == 0) { av[row] = ra; bv[row] = rb; }
}

// ---------------------------------------------------------------------------
// Decoder: out[i,j] = sigmoid( zw[i,:] . z[j,:] + a[i] + b[j] ).
// Gram GEMM C = ZW @ Z^T via WMMA; B fragment is a float2 from row j of z
// (since B[k][j] = z[j][k]). Fused rank-1 bias + sigmoid epilogue.
// One wave per 16x16 tile of the 1024x1024 output (4096 waves).
// ---------------------------------------------------------------------------
__global__ void decoder_wmma_kernel(const float* __restrict__ zw,
                                    const float* __restrict__ z,
                                    const float* __restrict__ av,
                                    const float* __restrict__ bv,
                                    float* __restrict__ out) {
    const int gwave = (blockIdx.x * blockDim.x + threadIdx.x) >> 5;
    const int lane  = threadIdx.x & 31;
    const int tm = gwave >> 6;        // 1024/16 = 64 tiles per dim
    const int tn = gwave & 63;
    const int lo = lane & 15;
    const int hi = lane >> 4;

    const float* arow = zw + (size_t)(tm * 16 + lo) * DD + 2 * hi;
    const float* brow = z  + (size_t)(tn * 16 + lo) * DD + 2 * hi;

    v8f acc = {};
    #pragma unroll 8
    for (int kb = 0; kb < DD; kb += 4) {
        v2f af = *(const v2f*)(arow + kb);
        v2f bf = *(const v2f*)(brow + kb);
        acc = __builtin_amdgcn_wmma_f32_16x16x4_f32(
            false, af, false, bf, (short)0, acc, false, false);
    }

    const int row0 = tm * 16 + hi * 8;
    const int col  = tn * 16 + lo;
    const float bj = bv[col];
    float* orow = out + (size_t)row0 * NN + col;
    #pragma unroll
    for (int g = 0; g < 8; ++g) {
        float s = acc[g] + av[row0 + g] + bj;
        orow[(size_t)g * NN] = 1.0f / (1.0f + __expf(-s));
    }
}

// ---------------------------------------------------------------------------
// Launch. Inputs (setup_inputs order):
//   0 inputs (1024x512 f32), 1 adj (1024x1024 f32), 2 weight (512x128 f32),
//   3 weight_two (256x128 f32), 4 weight_three (256 f32),
//   5/6 train edges (unused by the reference output).
// Workspace: x, z, zw (each 1024x128 f32), v (256), a (1024), b (1024).
// ---------------------------------------------------------------------------
extern "C" void kernel_launch(void* const* d_in, const int* in_sizes, int n_in,
                              void* d_out, int out_size, void* d_ws, size_t ws_size,
                              hipStream_t stream) {
    (void)in_sizes; (void)n_in; (void)out_size; (void)ws_size;

    const float* inputs = (const float*)d_in[0];
    const float* adj    = (const float*)d_in[1];
    const float* weight = (const float*)d_in[2];
    const float* w2     = (const float*)d_in[3];
    const float* w3     = (const float*)d_in[4];
    float* out = (float*)d_out;

    float* ws = (float*)d_ws;
    float* x  = ws;                 // 1024*128
    float* z  = x  + NN * DD;       // 1024*128
    float* zw = z  + NN * DD;       // 1024*128
    float* v  = zw + NN * DD;       // 256
    float* av = v  + 2 * DD;        // 1024
    float* bv = av + NN;            // 1024

    // v = W2 @ w3[:128]
    compute_v_kernel<<<1, 256, 0, stream>>>(w2, w3, v);

    // x = inputs @ weight   (1024x512 @ 512x128): 512 tiles, 4 waves/block
    gemm_f32_wmma<<<128, 128, 0, stream>>>(inputs, weight, x, NN, DD, DIN);

    // z = adj @ x           (1024x1024 @ 1024x128): 512 tiles
    gemm_f32_wmma<<<128, 128, 0, stream>>>(adj, x, z, NN, DD, NN);

    // per-row: zw, a, b     (1024 waves, 8 waves/block)
    row_stats_kernel<<<NN / 8, 256, 0, stream>>>(z, v, w3, zw, av, bv);

    // out = sigmoid(zw @ z^T + a[i] + b[j])  (4096 tiles, 8 waves/block)
    decoder_wmma_kernel<<<512, 256, 0, stream>>>(zw, z, av, bv, out);
}